// StrongFormPhysicsLoss_29669634081204
// MI455X (gfx1250) — compile-verified
//
#include <hip/hip_runtime.h>
#include <math.h>

typedef __attribute__((ext_vector_type(2))) float v2f;
typedef __attribute__((ext_vector_type(8))) float v8f;

// ---------------------------------------------------------------------------
// Wave32 sum reduction via V_WMMA_F32_16X16X4_F32.
// A (16x4): both A VGPRs = s  -> A[m][0..1]=s_m, A[m][2..3]=s_{m+16}
// B (4x16): all ones.
// D[m][n] = sum_k A[m][k] = 2*(s_m + s_{m+16})  (same for all n)
// Lane-local sum of the 8 D VGPRs gives 2*sum(s_0..7 + s_16..23) on lanes<16
// and 2*sum(s_8..15 + s_24..31) on lanes>=16; one shfl_xor(16) + *0.5 -> total.
// Requires EXEC all-ones: callers keep every lane alive (no divergent return).
// ---------------------------------------------------------------------------
__device__ __forceinline__ float wave_sum_wmma(float s) {
  v2f a; a[0] = s;    a[1] = s;
  v2f b; b[0] = 1.0f; b[1] = 1.0f;
  v8f c = {};
  v8f d = __builtin_amdgcn_wmma_f32_16x16x4_f32(
      /*neg_a=*/false, a, /*neg_b=*/false, b,
      /*c_mod=*/(short)0, c, /*reuse_a=*/false, /*reuse_b=*/false);
  float t = d[0] + d[1] + d[2] + d[3] + d[4] + d[5] + d[6] + d[7];
  t += __shfl_xor(t, 16, 32);
  return 0.5f * t;
}

__device__ __forceinline__ float wave_max32(float v) {
  #pragma unroll
  for (int o = 16; o > 0; o >>= 1) v = fmaxf(v, __shfl_xor(v, o, 32));
  return v;
}

// atomic max for non-negative floats via uint bit-pattern ordering
__device__ __forceinline__ void atomicMaxNonNeg(float* p, float v) {
  atomicMax((unsigned int*)p, __float_as_uint(v));
}

// scalar slot indices in workspace
#define S_KIN   0   // sum r_kin^2
#define S_CE    1   // sum (eps_ag - eps_fd)^2
#define S_CK    2   // sum (kappa_ag - kappa_fd)^2
#define S_QMAX  3   // max |elem_load|
#define S_LMAX  4   // max L
#define S_LSUM  5   // sum L
#define S_FORCE 6   // sum_{free_d} |F_int+F_ext|^2
#define S_MOM   7   // sum_{free_r} |M_int|^2
#define S_NEU   8   // sum_{pin}    |M_int|^2
#define S_FE    9   // sum_{free_d} |F_ext|^2
#define S_CNTD  10
#define S_CNTR  11
#define S_CNTP  12
#define NSCAL   16

__global__ void zero_ws_kernel(float* ws, long long n) {
  long long i = (long long)blockIdx.x * blockDim.x + threadIdx.x;
  long long stride = (long long)gridDim.x * blockDim.x;
  for (; i < n; i += stride) ws[i] = 0.0f;
}

// ---------------------------------------------------------------------------
// Element pass: per-element beam mechanics, atomic scatter to node
// accumulators, WMMA-reduced element-level sums + shuffle maxes.
// ---------------------------------------------------------------------------
__global__ void __launch_bounds__(256)
elem_kernel(const float* __restrict__ pred,
            const float* __restrict__ gux,
            const float* __restrict__ guz,
            const float* __restrict__ gphi,
            const float* __restrict__ pE,
            const float* __restrict__ pA,
            const float* __restrict__ pI,
            const float* __restrict__ pL,
            const float* __restrict__ dir,
            const float* __restrict__ eload,
            const int*   __restrict__ conn,
            float* __restrict__ accF,   // F_int + F_ext per node (3 floats)
            float* __restrict__ accFe,  // F_ext per node
            float* __restrict__ accM,   // M_int per node
            float* __restrict__ scal,
            int nElems)
{
  int e = blockIdx.x * blockDim.x + threadIdx.x;
  const bool ok = (e < nElems);
  const int ei = ok ? e : 0;

  // streamed element data
  const int   i   = conn[2 * ei + 0];
  const int   j   = conn[2 * ei + 1];
  const float xh0 = dir[3 * ei + 0];
  const float xh1 = dir[3 * ei + 1];
  const float xh2 = dir[3 * ei + 2];
  const float E   = pE[ei];
  const float EA  = E * pA[ei];
  const float EI  = E * pI[ei];
  const float L   = pL[ei];
  const float q0  = eload[3 * ei + 0];
  const float q1  = eload[3 * ei + 1];
  const float q2  = eload[3 * ei + 2];

  // local axes (matches jnp reference, clip(norm, 1e-8))
  const bool par = fabsf(xh1) > 0.99f;
  const float r1 = par ? 0.0f : 1.0f;
  const float r2 = par ? 1.0f : 0.0f;
  // z = cross(x, ref)
  float zx = xh1 * r2 - xh2 * r1;
  float zy = xh2 * 0.0f - xh0 * r2;
  float zz = xh0 * r1 - xh1 * 0.0f;
  {
    float inv = 1.0f / fmaxf(sqrtf(zx * zx + zy * zy + zz * zz), 1e-8f);
    zx *= inv; zy *= inv; zz *= inv;
  }
  // y = cross(z, x)
  float yx = zy * xh2 - zz * xh1;
  float yy = zz * xh0 - zx * xh2;
  float yz = zx * xh1 - zy * xh0;
  {
    float inv = 1.0f / fmaxf(sqrtf(yx * yx + yy * yy + yz * yz), 1e-8f);
    yx *= inv; yy *= inv; yz *= inv;
  }

  // gathered node data (L2-resident)
  const float uxi  = pred[3 * i + 0], uzi = pred[3 * i + 1], phii = pred[3 * i + 2];
  const float uxj  = pred[3 * j + 0], uzj = pred[3 * j + 1], phij = pred[3 * j + 2];

  const float dux = uxj - uxi;
  const float duz = uzj - uzi;
  const float du_ax = dux * xh0 + duz * xh2;
  const float du_tr = dux * zx + duz * zz;

  const float invL = 1.0f / L;
  const float eps_fd   = du_ax * invL;
  const float N_fd     = EA * eps_fd;
  const float kappa_fd = (phij - phii) * invL;
  const float EIL  = EI * invL;
  const float EIL2 = EIL * invL;
  const float EIL3 = EIL2 * invL;
  const float V_fd = 12.0f * EIL3 * du_tr - 6.0f * EIL2 * (phii + phij);
  const float M_yi = 6.0f * EIL2 * du_tr - EIL * (4.0f * phii + 2.0f * phij);
  const float M_yj = 6.0f * EIL2 * du_tr - EIL * (2.0f * phii + 4.0f * phij);

  // autograd-style strains from gradient fields
  const float gdi_ux  = gux[3 * i] * xh0 + gux[3 * i + 1] * xh1 + gux[3 * i + 2] * xh2;
  const float gdj_ux  = gux[3 * j] * xh0 + gux[3 * j + 1] * xh1 + gux[3 * j + 2] * xh2;
  const float gdi_uz  = guz[3 * i] * xh0 + guz[3 * i + 1] * xh1 + guz[3 * i + 2] * xh2;
  const float gdj_uz  = guz[3 * j] * xh0 + guz[3 * j + 1] * xh1 + guz[3 * j + 2] * xh2;
  const float gdi_phi = gphi[3 * i] * xh0 + gphi[3 * i + 1] * xh1 + gphi[3 * i + 2] * xh2;
  const float gdj_phi = gphi[3 * j] * xh0 + gphi[3 * j + 1] * xh1 + gphi[3 * j + 2] * xh2;
  const float eps_ag   = 0.5f * ((xh0 * gdi_ux + xh2 * gdi_uz) + (xh0 * gdj_ux + xh2 * gdj_uz));
  const float kappa_ag = 0.5f * (gdi_phi + gdj_phi);

  // edge force/moment vectors + distributed load halves
  const float Fx = N_fd * xh0 + V_fd * zx;
  const float Fy = N_fd * xh1 + V_fd * zy;
  const float Fz = N_fd * xh2 + V_fd * zz;
  const float hL = 0.5f * L;
  const float Hx = q0 * hL, Hy = q1 * hL, Hz = q2 * hL;

  if (ok) {
    // accF = F_int + F_ext
    atomicAdd(&accF[3 * i + 0], Fx + Hx);
    atomicAdd(&accF[3 * i + 1], Fy + Hy);
    atomicAdd(&accF[3 * i + 2], Fz + Hz);
    atomicAdd(&accF[3 * j + 0], -Fx + Hx);
    atomicAdd(&accF[3 * j + 1], -Fy + Hy);
    atomicAdd(&accF[3 * j + 2], -Fz + Hz);
    // accFe = F_ext only (for F_char)
    atomicAdd(&accFe[3 * i + 0], Hx);
    atomicAdd(&accFe[3 * i + 1], Hy);
    atomicAdd(&accFe[3 * i + 2], Hz);
    atomicAdd(&accFe[3 * j + 0], Hx);
    atomicAdd(&accFe[3 * j + 1], Hy);
    atomicAdd(&accFe[3 * j + 2], Hz);
    // accM = M_int
    atomicAdd(&accM[3 * i + 0], M_yi * yx);
    atomicAdd(&accM[3 * i + 1], M_yi * yy);
    atomicAdd(&accM[3 * i + 2], M_yi * yz);
    atomicAdd(&accM[3 * j + 0], M_yj * yx);
    atomicAdd(&accM[3 * j + 1], M_yj * yy);
    atomicAdd(&accM[3 * j + 2], M_yj * yz);
  }

  // element-level residual sums (zero-masked, all lanes stay live for WMMA)
  const float r_kin = 0.5f * (phii + phij) - du_tr * invL;
  const float de    = eps_ag - eps_fd;
  const float dk    = kappa_ag - kappa_fd;
  float s_kin = ok ? r_kin * r_kin : 0.0f;
  float s_ce  = ok ? de * de       : 0.0f;
  float s_ck  = ok ? dk * dk       : 0.0f;
  float s_L   = ok ? L             : 0.0f;
  float m_q   = ok ? fmaxf(fabsf(q0), fmaxf(fabsf(q1), fabsf(q2))) : 0.0f;
  float m_L   = ok ? L             : 0.0f;

  s_kin = wave_sum_wmma(s_kin);
  s_ce  = wave_sum_wmma(s_ce);
  s_ck  = wave_sum_wmma(s_ck);
  s_L   = wave_sum_wmma(s_L);
  m_q   = wave_max32(m_q);
  m_L   = wave_max32(m_L);

  __shared__ float red[8][6];
  const int wave = threadIdx.x >> 5;
  const int lane = threadIdx.x & 31;
  if (lane == 0) {
    red[wave][0] = s_kin; red[wave][1] = s_ce; red[wave][2] = s_ck;
    red[wave][3] = s_L;   red[wave][4] = m_q;  red[wave][5] = m_L;
  }
  __syncthreads();
  if (threadIdx.x == 0) {
    const int nw = (blockDim.x + 31) >> 5;
    float a0 = 0, a1 = 0, a2 = 0, a3 = 0, a4 = 0, a5 = 0;
    for (int w = 0; w < nw; ++w) {
      a0 += red[w][0]; a1 += red[w][1]; a2 += red[w][2]; a3 += red[w][3];
      a4 = fmaxf(a4, red[w][4]); a5 = fmaxf(a5, red[w][5]);
    }
    atomicAdd(&scal[S_KIN], a0);
    atomicAdd(&scal[S_CE], a1);
    atomicAdd(&scal[S_CK], a2);
    atomicAdd(&scal[S_LSUM], a3);
    atomicMaxNonNeg(&scal[S_QMAX], a4);
    atomicMaxNonNeg(&scal[S_LMAX], a5);
  }
}

// ---------------------------------------------------------------------------
// Node pass: masked unnormalized sums + counts, WMMA-reduced.
// ---------------------------------------------------------------------------
__global__ void __launch_bounds__(256)
node_kernel(const float* __restrict__ accF,
            const float* __restrict__ accFe,
            const float* __restrict__ accM,
            const float* __restrict__ bcd,
            const float* __restrict__ bcr,
            float* __restrict__ scal,
            int nNodes)
{
  int n = blockIdx.x * blockDim.x + threadIdx.x;
  const bool ok = (n < nNodes);
  const int ni = ok ? n : 0;

  const float fx = accF[3 * ni], fy = accF[3 * ni + 1], fz = accF[3 * ni + 2];
  const float ex = accFe[3 * ni], ey = accFe[3 * ni + 1], ez = accFe[3 * ni + 2];
  const float mx = accM[3 * ni], my = accM[3 * ni + 1], mz = accM[3 * ni + 2];
  const float f2 = fx * fx + fy * fy + fz * fz;
  const float e2 = ex * ex + ey * ey + ez * ez;
  const float m2 = mx * mx + my * my + mz * mz;

  const float bd = bcd[ni], br = bcr[ni];
  const bool free_d = ok && (bd < 0.5f);
  const bool free_r = ok && (br < 0.5f);
  const bool pin    = ok && (bd > 0.5f) && (br < 0.5f);

  float sF  = free_d ? f2 : 0.0f;
  float sFe = free_d ? e2 : 0.0f;
  float sM  = free_r ? m2 : 0.0f;
  float sN  = pin    ? m2 : 0.0f;
  float cD  = free_d ? 1.0f : 0.0f;
  float cR  = free_r ? 1.0f : 0.0f;
  float cP  = pin    ? 1.0f : 0.0f;

  sF  = wave_sum_wmma(sF);
  sFe = wave_sum_wmma(sFe);
  sM  = wave_sum_wmma(sM);
  sN  = wave_sum_wmma(sN);
  cD  = wave_sum_wmma(cD);
  cR  = wave_sum_wmma(cR);
  cP  = wave_sum_wmma(cP);

  __shared__ float red[8][7];
  const int wave = threadIdx.x >> 5;
  const int lane = threadIdx.x & 31;
  if (lane == 0) {
    red[wave][0] = sF; red[wave][1] = sFe; red[wave][2] = sM; red[wave][3] = sN;
    red[wave][4] = cD; red[wave][5] = cR;  red[wave][6] = cP;
  }
  __syncthreads();
  if (threadIdx.x == 0) {
    const int nw = (blockDim.x + 31) >> 5;
    float a[7] = {0, 0, 0, 0, 0, 0, 0};
    for (int w = 0; w < nw; ++w)
      for (int k = 0; k < 7; ++k) a[k] += red[w][k];
    atomicAdd(&scal[S_FORCE], a[0]);
    atomicAdd(&scal[S_FE], a[1]);
    atomicAdd(&scal[S_MOM], a[2]);
    atomicAdd(&scal[S_NEU], a[3]);
    atomicAdd(&scal[S_CNTD], a[4]);
    atomicAdd(&scal[S_CNTR], a[5]);
    atomicAdd(&scal[S_CNTP], a[6]);
  }
}

__global__ void finalize_kernel(const float* __restrict__ scal,
                                float* __restrict__ out, float nElems)
{
  if (threadIdx.x != 0 || blockIdx.x != 0) return;
  const float cD = fmaxf(scal[S_CNTD], 1.0f) * 3.0f;
  const float cR = fmaxf(scal[S_CNTR], 1.0f) * 3.0f;
  const float cP = fmaxf(scal[S_CNTP], 1.0f) * 3.0f;

  const float F_char = fmaxf(sqrtf(scal[S_FE] / cD), 1.0f);
  const float M_char = fmaxf(scal[S_QMAX] * scal[S_LMAX] * scal[S_LSUM] * 0.125f, 1.0f);

  const float L_force   = scal[S_FORCE] / (F_char * F_char * cD);
  const float L_moment  = scal[S_MOM] / (M_char * M_char * cR);
  const float L_neumann = scal[S_NEU] / (M_char * M_char * cP);
  const float L_kin     = scal[S_KIN] / nElems;
  const float L_consist = (scal[S_CE] + scal[S_CK]) / nElems;

  // W_FORCE=1, W_MOMENT=1, W_KIN=0.1, W_NEUMANN=1, W_CONSISTENCY=1
  out[0] = L_force + L_moment + L_neumann + 0.1f * L_kin + L_consist;
}

extern "C" void kernel_launch(void* const* d_in, const int* in_sizes, int n_in,
                              void* d_out, int out_size, void* d_ws, size_t ws_size,
                              hipStream_t stream) {
  (void)n_in; (void)out_size; (void)ws_size;
  const float* pred  = (const float*)d_in[0];
  const float* gux   = (const float*)d_in[1];
  const float* guz   = (const float*)d_in[2];
  const float* gphi  = (const float*)d_in[3];
  const float* pE    = (const float*)d_in[4];
  const float* pA    = (const float*)d_in[5];
  const float* pI    = (const float*)d_in[6];
  const float* pL    = (const float*)d_in[7];
  const float* dir   = (const float*)d_in[8];
  const float* eload = (const float*)d_in[9];
  const float* bcd   = (const float*)d_in[10];
  const float* bcr   = (const float*)d_in[11];
  const int*   conn  = (const int*)d_in[12];

  const int nNodes = in_sizes[0] / 3;
  const int nElems = in_sizes[4];

  float* accF  = (float*)d_ws;              // 3*nNodes
  float* accFe = accF + (size_t)3 * nNodes; // 3*nNodes
  float* accM  = accFe + (size_t)3 * nNodes;// 3*nNodes
  float* scal  = accM + (size_t)3 * nNodes; // NSCAL

  const long long zN = (long long)9 * nNodes + NSCAL;
  int zb = (int)((zN + 255) / 256);
  if (zb > 16384) zb = 16384;
  zero_ws_kernel<<<zb, 256, 0, stream>>>((float*)d_ws, zN);

  elem_kernel<<<(nElems + 255) / 256, 256, 0, stream>>>(
      pred, gux, guz, gphi, pE, pA, pI, pL, dir, eload, conn,
      accF, accFe, accM, scal, nElems);

  node_kernel<<<(nNodes + 255) / 256, 256, 0, stream>>>(
      accF, accFe, accM, bcd, bcr, scal, nNodes);

  finalize_kernel<<<1, 32, 0, stream>>>(scal, (float*)d_out, (float)nElems);
}